// RelPartialLearnableMultiHeadAttn_51917564674334
// MI455X (gfx1250) — compile-verified
//
#include <hip/hip_runtime.h>

typedef __attribute__((ext_vector_type(2))) float v2f;
typedef __attribute__((ext_vector_type(8))) float v8f;

#define QLEN 512
#define MLEN 512
#define KLEN 1024
#define BSZ  16
#define DM   1024
#define NH   16
#define DH   64
#define NDH  (NH*DH)     // 1024
#define QKV3 (3*NDH)     // 3072
#define SCALE 0.125f     // 1/sqrt(64)
#define LN_EPS 1e-5f

// f32 WMMA: D(16x16 f32) = A(16x4 f32) * B(4x16 f32) + C
// A/B frag: lane l holds (m|n = l&15, k = 2*(l>>4) + {0,1}) -> v2f
// C/D frag: lane l holds (M = p + 8*(l>>4), N = l&15) for p = 0..7 -> v8f
__device__ __forceinline__ v8f wmma4(v2f a, v2f b, v8f c) {
  return __builtin_amdgcn_wmma_f32_16x16x4_f32(
      /*neg_a=*/false, a, /*neg_b=*/false, b,
      /*c_mod=*/(short)0, c, /*reuse_a=*/false, /*reuse_b=*/false);
}

// ---------------------------------------------------------------------------
// Kernel 1: w_heads = concat(mems, w) @ Wqkv, scattered into q/k/v buffers.
// One wave per block, 16(M) x 64(N) per wave. grid = (3072/64, 16384/16).
// q buffer:  [QLEN*BSZ, NDH]   (only rows >= MLEN*BSZ produce q)
// k,v:       [KLEN*BSZ, NDH]
// ---------------------------------------------------------------------------
__global__ void qkv_gemm_kernel(const float* __restrict__ w,
                                const float* __restrict__ mems,
                                const float* __restrict__ Wqkv,
                                float* __restrict__ qb_,
                                float* __restrict__ kb_,
                                float* __restrict__ vb_) {
  const int lane = threadIdx.x & 31;
  const int m    = lane & 15;
  const int kp   = (lane >> 4) << 1;
  const int mhi  = (lane >> 4) << 3;
  const int n0   = blockIdx.x * 64;
  const int tm   = blockIdx.y;
  // q columns for pure-mem row tiles are never used downstream: skip.
  if (n0 < NDH && tm < (MLEN * BSZ / 16)) return;
  const int row = tm * 16 + m;
  const float* Arow = (row < MLEN * BSZ)
                          ? (mems + (size_t)row * DM)
                          : (w + (size_t)(row - MLEN * BSZ) * DM);
  const int nc = n0 + m;
  v8f a0 = {}, a1 = {}, a2 = {}, a3 = {};
  for (int k0 = 0; k0 < DM; k0 += 4) {
    v2f a = *(const v2f*)(Arow + k0 + kp);
    const float* B0 = Wqkv + (size_t)(k0 + kp) * QKV3;
    const float* B1 = B0 + QKV3;
    v2f b;
    b.x = B0[nc];      b.y = B1[nc];      a0 = wmma4(a, b, a0);
    b.x = B0[nc + 16]; b.y = B1[nc + 16]; a1 = wmma4(a, b, a1);
    b.x = B0[nc + 32]; b.y = B1[nc + 32]; a2 = wmma4(a, b, a2);
    b.x = B0[nc + 48]; b.y = B1[nc + 48]; a3 = wmma4(a, b, a3);
  }
  v8f accs[4] = {a0, a1, a2, a3};
#pragma unroll
  for (int t = 0; t < 4; ++t) {
    const int col = n0 + t * 16 + m;   // whole 64-col block stays in one segment
#pragma unroll
    for (int p = 0; p < 8; ++p) {
      const int r = tm * 16 + p + mhi;
      const float val = accs[t][p];
      if (col < NDH) {
        if (r >= MLEN * BSZ) qb_[(size_t)(r - MLEN * BSZ) * NDH + col] = val;
      } else if (col < 2 * NDH) {
        kb_[(size_t)r * NDH + (col - NDH)] = val;
      } else {
        vb_[(size_t)r * NDH + (col - 2 * NDH)] = val;
      }
    }
  }
}

// ---------------------------------------------------------------------------
// Kernel 2: r_head_k = r @ Wr  -> [KLEN, NDH].  grid = (1024/64, 1024/16).
// ---------------------------------------------------------------------------
__global__ void rk_gemm_kernel(const float* __restrict__ r,
                               const float* __restrict__ Wr,
                               float* __restrict__ rk) {
  const int lane = threadIdx.x & 31;
  const int m    = lane & 15;
  const int kp   = (lane >> 4) << 1;
  const int mhi  = (lane >> 4) << 3;
  const int n0   = blockIdx.x * 64;
  const int tm   = blockIdx.y;
  const float* Arow = r + (size_t)(tm * 16 + m) * DM;
  const int nc = n0 + m;
  v8f a0 = {}, a1 = {}, a2 = {}, a3 = {};
  for (int k0 = 0; k0 < DM; k0 += 4) {
    v2f a = *(const v2f*)(Arow + k0 + kp);
    const float* B0 = Wr + (size_t)(k0 + kp) * NDH;
    const float* B1 = B0 + NDH;
    v2f b;
    b.x = B0[nc];      b.y = B1[nc];      a0 = wmma4(a, b, a0);
    b.x = B0[nc + 16]; b.y = B1[nc + 16]; a1 = wmma4(a, b, a1);
    b.x = B0[nc + 32]; b.y = B1[nc + 32]; a2 = wmma4(a, b, a2);
    b.x = B0[nc + 48]; b.y = B1[nc + 48]; a3 = wmma4(a, b, a3);
  }
  v8f accs[4] = {a0, a1, a2, a3};
#pragma unroll
  for (int t = 0; t < 4; ++t)
#pragma unroll
    for (int p = 0; p < 8; ++p)
      rk[(size_t)(tm * 16 + p + mhi) * NDH + n0 + t * 16 + m] = accs[t][p];
}

// ---------------------------------------------------------------------------
// Kernel 3: fused attention per (q-tile 16, b, h). 4 waves / block.
// Stage 1: AC = (q+rwb)·k^T and BD_pre = (q+rrb)·r_head_k^T, 16x1024 each in LDS
// Stage 2: rel-shifted + masked + scaled softmax (BD[i][j]=BD_pre[i][j+511-i])
// Stage 3: attn_vec = P @ V (16x64) -> av buffer [QLEN*BSZ, NDH]
// ---------------------------------------------------------------------------
__global__ void attn_kernel(const float* __restrict__ qbuf,
                            const float* __restrict__ kbuf,
                            const float* __restrict__ vbuf,
                            const float* __restrict__ rk,
                            const float* __restrict__ rwb,
                            const float* __restrict__ rrb,
                            float* __restrict__ av) {
  extern __shared__ float lds[];
  float* sA = lds;               // [16][KLEN] : AC scores, then probabilities
  float* sB = lds + 16 * KLEN;   // [16][KLEN] : BD_pre (un-shifted)
  const int lane = threadIdx.x & 31;
  const int wid  = threadIdx.x >> 5;      // 0..3
  const int m    = lane & 15;
  const int kp   = (lane >> 4) << 1;
  const int mhi  = (lane >> 4) << 3;
  const int qt   = blockIdx.x;            // 0..31
  const int b    = blockIdx.y;
  const int h    = blockIdx.z;

  // Preload per-lane q fragments with both bias variants (K=64 -> 16 k-steps).
  const int igl = qt * 16 + m;
  const float* qrow = qbuf + (size_t)(igl * BSZ + b) * NDH + h * DH;
  const float* rwbh = rwb + h * DH;
  const float* rrbh = rrb + h * DH;
  v2f qaf[16], qbf[16];
#pragma unroll
  for (int kt = 0; kt < 16; ++kt) {
    const int d = kt * 4 + kp;
    v2f qv = *(const v2f*)(qrow + d);
    qaf[kt] = qv + *(const v2f*)(rwbh + d);
    qbf[kt] = qv + *(const v2f*)(rrbh + d);
  }

  // Stage 1: each wave does 16 of the 64 key tiles (AC and BD together).
  const size_t kcb = (size_t)b * NDH + h * DH;
  for (int jt = 0; jt < 16; ++jt) {
    const int j0   = (jt * 4 + wid) * 16;
    const int jcol = j0 + m;                              // this lane's B column
    const float* kc = kbuf + (size_t)jcol * BSZ * NDH + kcb;
    const float* rc = rk + (size_t)jcol * NDH + h * DH;
    v8f aAC = {}, aBD = {};
#pragma unroll
    for (int kt = 0; kt < 16; ++kt) {
      const int d = kt * 4 + kp;
      v2f bk = *(const v2f*)(kc + d);
      v2f br = *(const v2f*)(rc + d);
      aAC = wmma4(qaf[kt], bk, aAC);
      aBD = wmma4(qbf[kt], br, aBD);
    }
#pragma unroll
    for (int p = 0; p < 8; ++p) {
      sA[(p + mhi) * KLEN + j0 + m] = aAC[p];
      sB[(p + mhi) * KLEN + j0 + m] = aBD[p];
    }
  }
  __syncthreads();

  // Stage 2: softmax; 4 rows per wave, 32 lanes per row.
#pragma unroll
  for (int rr = 0; rr < 4; ++rr) {
    const int i     = wid * 4 + rr;
    const int ig    = qt * 16 + i;
    const int jmax  = ig + MLEN;          // inclusive; j>jmax is masked (-inf)
    const int shift = QLEN - 1 - ig;      // rel_shift read offset
    float sc[32];
    float mx = -3.0e38f;
#pragma unroll
    for (int t = 0; t < 32; ++t) {
      const int j = lane + t * 32;
      float s = (j <= jmax)
                    ? (sA[i * KLEN + j] + sB[i * KLEN + j + shift]) * SCALE
                    : -3.0e38f;
      sc[t] = s;
      mx = fmaxf(mx, s);
    }
    for (int off = 16; off > 0; off >>= 1) mx = fmaxf(mx, __shfl_xor(mx, off, 32));
    float sum = 0.f;
#pragma unroll
    for (int t = 0; t < 32; ++t) {
      const int j = lane + t * 32;
      const float e = (j <= jmax) ? __expf(sc[t] - mx) : 0.f;
      sc[t] = e;
      sum += e;
    }
    for (int off = 16; off > 0; off >>= 1) sum += __shfl_xor(sum, off, 32);
    const float inv = 1.f / sum;
#pragma unroll
    for (int t = 0; t < 32; ++t) sA[i * KLEN + lane + t * 32] = sc[t] * inv;
  }
  __syncthreads();

  // Stage 3: attn_vec tile = P(16x1024, LDS) @ V(1024x64); wave -> 16 d-cols.
  const int d0 = wid * 16;
  const size_t vb = (size_t)b * NDH + h * DH + d0 + m;
  v8f acc = {};
  for (int k0 = 0; k0 < KLEN; k0 += 4) {
    v2f a = *(const v2f*)(&sA[m * KLEN + k0 + kp]);
    v2f bv;
    bv.x = vbuf[(size_t)(k0 + kp) * BSZ * NDH + vb];
    bv.y = vbuf[(size_t)(k0 + kp + 1) * BSZ * NDH + vb];
    acc = wmma4(a, bv, acc);
  }
#pragma unroll
  for (int p = 0; p < 8; ++p) {
    const int ig = qt * 16 + p + mhi;
    av[(size_t)(ig * BSZ + b) * NDH + h * DH + d0 + m] = acc[p];
  }
}

// ---------------------------------------------------------------------------
// Kernel 4: out = LayerNorm(w + av @ Wo). 16 rows / block, 4 waves.
// Each wave computes 16 x 256 of the projection into LDS, then 4 rows of LN.
// ---------------------------------------------------------------------------
__global__ void proj_ln_kernel(const float* __restrict__ av,
                               const float* __restrict__ Wo,
                               const float* __restrict__ wres,
                               const float* __restrict__ gamma,
                               const float* __restrict__ beta,
                               float* __restrict__ out) {
  extern __shared__ float sO[];  // [16][NDH]
  const int lane = threadIdx.x & 31;
  const int wid  = threadIdx.x >> 5;
  const int m    = lane & 15;
  const int kp   = (lane >> 4) << 1;
  const int mhi  = (lane >> 4) << 3;
  const int r0   = blockIdx.x * 16;
  const float* Arow = av + (size_t)(r0 + m) * NDH;
#pragma unroll
  for (int ng = 0; ng < 4; ++ng) {
    const int n0 = wid * 256 + ng * 64;
    const int nc = n0 + m;
    v8f a0 = {}, a1 = {}, a2 = {}, a3 = {};
    for (int k0 = 0; k0 < DM; k0 += 4) {
      v2f a = *(const v2f*)(Arow + k0 + kp);
      const float* B0 = Wo + (size_t)(k0 + kp) * NDH;
      const float* B1 = B0 + NDH;
      v2f b;
      b.x = B0[nc];      b.y = B1[nc];      a0 = wmma4(a, b, a0);
      b.x = B0[nc + 16]; b.y = B1[nc + 16]; a1 = wmma4(a, b, a1);
      b.x = B0[nc + 32]; b.y = B1[nc + 32]; a2 = wmma4(a, b, a2);
      b.x = B0[nc + 48]; b.y = B1[nc + 48]; a3 = wmma4(a, b, a3);
    }
    v8f accs[4] = {a0, a1, a2, a3};
#pragma unroll
    for (int t = 0; t < 4; ++t)
#pragma unroll
      for (int p = 0; p < 8; ++p)
        sO[(p + mhi) * NDH + n0 + t * 16 + m] = accs[t][p];
  }
  __syncthreads();
#pragma unroll
  for (int rr = 0; rr < 4; ++rr) {
    const int i = wid * 4 + rr;
    const size_t rbase = (size_t)(r0 + i) * NDH;
    float s = 0.f, sq = 0.f;
    for (int t = 0; t < 32; ++t) {
      const int j = lane + t * 32;
      const float x = sO[i * NDH + j] + wres[rbase + j];
      s += x; sq += x * x;
    }
    for (int off = 16; off > 0; off >>= 1) {
      s  += __shfl_xor(s, off, 32);
      sq += __shfl_xor(sq, off, 32);
    }
    const float mu   = s * (1.f / (float)DM);
    const float var  = sq * (1.f / (float)DM) - mu * mu;
    const float rstd = rsqrtf(var + LN_EPS);
    for (int t = 0; t < 32; ++t) {
      const int j = lane + t * 32;
      const float x = sO[i * NDH + j] + wres[rbase + j];
      out[rbase + j] = (x - mu) * rstd * gamma[j] + beta[j];
    }
  }
}

// ---------------------------------------------------------------------------
extern "C" void kernel_launch(void* const* d_in, const int* in_sizes, int n_in,
                              void* d_out, int out_size, void* d_ws, size_t ws_size,
                              hipStream_t stream) {
  (void)in_sizes; (void)n_in; (void)out_size; (void)ws_size;
  const float* w     = (const float*)d_in[0];   // [512,16,1024]
  const float* r     = (const float*)d_in[1];   // [1024,1024]
  const float* rwb   = (const float*)d_in[2];   // [16,64]
  const float* rrb   = (const float*)d_in[3];   // [16,64]
  const float* mems  = (const float*)d_in[4];   // [512,16,1024]
  // d_in[5] attn_mask is analytic (j > i + MLEN): recomputed in-kernel.
  const float* Wqkv  = (const float*)d_in[6];   // [1024,3072]
  const float* Wr    = (const float*)d_in[7];   // [1024,1024]
  const float* Wo    = (const float*)d_in[8];   // [1024,1024]
  const float* gamma = (const float*)d_in[9];   // [1024]
  const float* beta  = (const float*)d_in[10];  // [1024]
  float* out = (float*)d_out;

  float* ws  = (float*)d_ws;
  float* qb_ = ws;                                   // 512*16*1024   (32 MB)
  float* kb_ = qb_ + (size_t)QLEN * BSZ * NDH;       // 1024*16*1024  (64 MB)
  float* vb_ = kb_ + (size_t)KLEN * BSZ * NDH;       // 1024*16*1024  (64 MB)
  float* rk  = vb_ + (size_t)KLEN * BSZ * NDH;       // 1024*1024     ( 4 MB)
  float* avb = rk  + (size_t)KLEN * NDH;             // 512*16*1024   (32 MB)

  qkv_gemm_kernel<<<dim3(QKV3 / 64, KLEN * BSZ / 16), 32, 0, stream>>>(
      w, mems, Wqkv, qb_, kb_, vb_);
  rk_gemm_kernel<<<dim3(NDH / 64, KLEN / 16), 32, 0, stream>>>(r, Wr, rk);
  attn_kernel<<<dim3(QLEN / 16, BSZ, NH), 128,
                2u * 16u * KLEN * sizeof(float), stream>>>(
      qb_, kb_, vb_, rk, rwb, rrb, avb);
  proj_ln_kernel<<<dim3(QLEN * BSZ / 16), 128, 16u * NDH * sizeof(float),
                   stream>>>(avb, Wo, w, gamma, beta, out);
}